// BatchedSharedLoRA_31284541784342
// MI455X (gfx1250) — compile-verified
//
#include <hip/hip_runtime.h>
#include <math.h>

typedef float v2f __attribute__((ext_vector_type(2)));
typedef float v8f __attribute__((ext_vector_type(8)));

// Problem constants (fixed by the reference harness)
constexpr int   NAD = 8;            // adapters
constexpr int   H   = 4096;         // hidden
constexpr int   R   = 64;           // lora rank
constexpr int   BS  = 2 * 2048;     // B*S flattened rows
constexpr float SCALING = 2.0f;
constexpr float EPS     = 1e-8f;

// ---------------------------------------------------------------------------
// Kernel 1: inter[n, bs, r] = sum_h x[bs, h] * A[n, h, r]
// One wave computes a 16(M) x 64(R) tile, K-loop over H in steps of 4 using
// V_WMMA_F32_16X16X4_F32. 2048 wave-tiles total -> 256 blocks of 8 waves.
// ---------------------------------------------------------------------------
__global__ __launch_bounds__(256) void lora_gemm1_kernel(
    const float* __restrict__ x,      // [BS, H]
    const float* __restrict__ A,      // [NAD, H, R]
    float* __restrict__ inter)        // [NAD, BS, R]
{
    const int wave = threadIdx.x >> 5;
    const int lane = threadIdx.x & 31;
    const int half = lane >> 4;            // 0: lanes 0-15, 1: lanes 16-31
    const int lm   = lane & 15;

    const int tile = blockIdx.x * 8 + wave;   // 0 .. 2047
    const int n    = tile >> 8;               // adapter
    const int mb   = (tile & 255) << 4;       // row base in BS

    // A-operand (x): lane holds row mb+lm, K = h + 2*half + {0,1}
    const float* xrow = x + (size_t)(mb + lm) * H + 2 * half;
    const float* An   = A + (size_t)n * H * R;
    float*       In   = inter + (size_t)n * BS * R;

    v8f acc0 = {}, acc1 = {}, acc2 = {}, acc3 = {};

    for (int h = 0; h < H; h += 4) {
        if ((h & 127) == 0)
            __builtin_prefetch(xrow + h + 128, 0, 0);   // global_prefetch_b8

        v2f af = *(const v2f*)(xrow + h);

        // B-operand (lora_A): lane holds col (t*16+lm), K = 2*half + {0,1}
        const float* bp = An + (size_t)(h + 2 * half) * R + lm;
        v2f b0 = { bp[ 0], bp[R +  0] };
        v2f b1 = { bp[16], bp[R + 16] };
        v2f b2 = { bp[32], bp[R + 32] };
        v2f b3 = { bp[48], bp[R + 48] };

        acc0 = __builtin_amdgcn_wmma_f32_16x16x4_f32(false, af, false, b0, (short)0, acc0, false, false);
        acc1 = __builtin_amdgcn_wmma_f32_16x16x4_f32(false, af, false, b1, (short)0, acc1, false, false);
        acc2 = __builtin_amdgcn_wmma_f32_16x16x4_f32(false, af, false, b2, (short)0, acc2, false, false);
        acc3 = __builtin_amdgcn_wmma_f32_16x16x4_f32(false, af, false, b3, (short)0, acc3, false, false);
    }

    // C/D layout: VGPR i -> row (i + 8*half), col = lm (+ tile offset)
    #pragma unroll
    for (int i = 0; i < 8; ++i) {
        float* op = In + (size_t)(mb + i + 8 * half) * R + lm;
        op[ 0] = acc0[i];
        op[16] = acc1[i];
        op[32] = acc2[i];
        op[48] = acc3[i];
    }
}

// ---------------------------------------------------------------------------
// Kernel 2 (fused): upd = inter . B ; row-norm over H ; out = x + upd/||upd||*2
// One wave owns one 16-row tile over the full H. inter fragment (16x64)
// preloaded once into registers. Pass 1 accumulates sum-of-squares only,
// butterfly-reduce across the 16-lane M-groups, pass 2 recomputes (K=64 is
// cheap) and writes the final output exactly once.
// ---------------------------------------------------------------------------
__global__ __launch_bounds__(256) void lora_gemm2_fused_kernel(
    const float* __restrict__ x,      // [BS, H]
    const float* __restrict__ Bm,     // [NAD, R, H]
    const float* __restrict__ inter,  // [NAD, BS, R]
    float* __restrict__ out)          // [NAD, BS, H]
{
    const int wave = threadIdx.x >> 5;
    const int lane = threadIdx.x & 31;
    const int half = lane >> 4;
    const int lm   = lane & 15;

    const int tile = blockIdx.x * 8 + wave;   // 0 .. 2047
    const int n    = tile >> 8;
    const int mb   = (tile & 255) << 4;

    // Preload 16 A-fragments (full K=64) of inter for this wave's 16 rows.
    const float* In = inter + ((size_t)n * BS + mb + lm) * R + 2 * half;
    v2f af[16];
    #pragma unroll
    for (int kk = 0; kk < 16; ++kk)
        af[kk] = *(const v2f*)(In + 4 * kk);

    const float* Bn = Bm + (size_t)n * R * H;

    // ---- Pass 1: accumulate per-lane sum of squares over all H ----
    float ps[8] = {0.f, 0.f, 0.f, 0.f, 0.f, 0.f, 0.f, 0.f};
    for (int ct = 0; ct < H / 16; ++ct) {
        const int col = ct * 16 + lm;
        v8f acc = {};
        #pragma unroll
        for (int kk = 0; kk < 16; ++kk) {
            const float* bp = Bn + (size_t)(4 * kk + 2 * half) * H + col;
            v2f bf = { bp[0], bp[H] };
            acc = __builtin_amdgcn_wmma_f32_16x16x4_f32(false, af[kk], false, bf, (short)0, acc, false, false);
        }
        #pragma unroll
        for (int i = 0; i < 8; ++i)
            ps[i] += acc[i] * acc[i];
    }

    // Butterfly reduce within each 16-lane group (rows 0-7 in lanes 0-15,
    // rows 8-15 in lanes 16-31); masks < 16 keep the halves separate.
    #pragma unroll
    for (int msk = 1; msk < 16; msk <<= 1) {
        #pragma unroll
        for (int i = 0; i < 8; ++i)
            ps[i] += __shfl_xor(ps[i], msk, 32);
    }

    float sc[8];
    #pragma unroll
    for (int i = 0; i < 8; ++i)
        sc[i] = SCALING / (sqrtf(ps[i]) + EPS);

    // ---- Pass 2: recompute tiles, scale, add residual, write once ----
    const float* xb = x   + (size_t)mb * H;
    float*       ob = out + ((size_t)n * BS + mb) * H;
    for (int ct = 0; ct < H / 16; ++ct) {
        const int col = ct * 16 + lm;
        v8f acc = {};
        #pragma unroll
        for (int kk = 0; kk < 16; ++kk) {
            const float* bp = Bn + (size_t)(4 * kk + 2 * half) * H + col;
            v2f bf = { bp[0], bp[H] };
            acc = __builtin_amdgcn_wmma_f32_16x16x4_f32(false, af[kk], false, bf, (short)0, acc, false, false);
        }
        #pragma unroll
        for (int i = 0; i < 8; ++i) {
            const int ro = i + 8 * half;
            ob[(size_t)ro * H + col] = xb[(size_t)ro * H + col] + acc[i] * sc[i];
        }
    }
}

// ---------------------------------------------------------------------------
extern "C" void kernel_launch(void* const* d_in, const int* in_sizes, int n_in,
                              void* d_out, int out_size, void* d_ws, size_t ws_size,
                              hipStream_t stream) {
    (void)in_sizes; (void)n_in; (void)out_size; (void)ws_size;

    const float* x  = (const float*)d_in[0];   // [2,2048,4096]
    const float* A  = (const float*)d_in[1];   // [8,4096,64]
    const float* Bm = (const float*)d_in[2];   // [8,64,4096]
    float* out      = (float*)d_out;           // [8,2,2048,4096]
    float* inter    = (float*)d_ws;            // [8,4096,64] = 8 MB scratch

    // 2048 wave-tiles each, 8 waves (256 threads) per block.
    lora_gemm1_kernel<<<256, 256, 0, stream>>>(x, A, inter);
    lora_gemm2_fused_kernel<<<256, 256, 0, stream>>>(x, Bm, inter, out);
}